// M2A2_58291296141949
// MI455X (gfx1250) — compile-verified
//
#include <hip/hip_runtime.h>
#include <math.h>

// ---------------------------------------------------------------------------
// CDNA5 (gfx1250) f32 WMMA helpers.  V_WMMA_F32_16X16X4_F32:
//   A: 16x4 f32 (2 VGPR/lane), B: 4x16 f32 (2 VGPR/lane), C/D: 16x16 f32 (8).
// A-frag: lane holds A[M=lane%16][K = kb + (lane/16)*2 + {0,1}]
// B-frag: lane holds B[K = kb + (lane/16)*2 + {0,1}][N=lane%16]
// C/D   : lane,vgpr r holds D[M = r + 8*(lane/16)][N = lane%16]
// ---------------------------------------------------------------------------
typedef __attribute__((ext_vector_type(2))) float v2f;
typedef __attribute__((ext_vector_type(8))) float v8f;

__device__ __forceinline__ v8f wmma_f32(v2f a, v2f b, v8f c) {
  return __builtin_amdgcn_wmma_f32_16x16x4_f32(false, a, false, b,
                                               (short)0, c, false, false);
}

#define DIMC   256
#define HW     4096   // 64*64
#define TSTRIDE 131072LL   // 32*4096 (per-batch stride of t / q2 / out5)

// ---------------------------------------------------------------------------
// Kernel 0: repack conv weights OIHW (oc,ic,kh,kw) -> (oc, tap, ic) so the
// conv A-fragment (two k-consecutive ic for one oc) is one aligned b64 load.
// ---------------------------------------------------------------------------
__global__ void __launch_bounds__(256)
k_repack_w(const float* __restrict__ wc, float* __restrict__ wp) {
  const int idx = blockIdx.x * 256 + threadIdx.x;   // flat over [oc][tap][ic]
  const int ic  = idx & 255;
  const int tap = (idx >> 8) % 9;
  const int oc  = idx / (9 * 256);
  wp[idx] = wc[(oc * 256 + ic) * 9 + tap];
}

// ---------------------------------------------------------------------------
// Kernel 1: dense 3x3 conv 256->256, pad 1, implicit GEMM with f32 WMMA.
// Grid: 1024 blocks = 4(n) * 4(oc-group of 64) * 64(h rows). Block: 128 (4 waves).
// Each wave: 16 oc x 64 pixels (one image row), K = 256ic * 9taps = 2304.
// LDS slab layout [row][col][icl] (ic-fastest): both WMMA fragments become
// single aligned 64-bit loads (ds_load_b64 for B, global b64 for A).
// ---------------------------------------------------------------------------
__global__ void __launch_bounds__(128)
k_conv3x3(const float* __restrict__ x, const float* __restrict__ wp,
          const float* __restrict__ bc, float* __restrict__ y0) {
  __shared__ float slab[3 * 66 * 16];             // [(row*66+col)*16 + icl]
  const int bid = blockIdx.x;
  const int n = bid >> 8, ocg = (bid >> 6) & 3, h = bid & 63;
  const int tid = threadIdx.x;
  const int lane = tid & 15, hh = (tid >> 4) & 1, wv = tid >> 5;
  const int ocbase = ocg * 64 + wv * 16;
  const int koff = hh * 2;                         // lane's k sub-offset

  v8f acc[4] = {};
  for (int icb = 0; icb < 16; ++icb) {
    __syncthreads();
    // stage: enumerate (icl, row, col) with col fastest -> coalesced global
    // reads; scatter-write the ic-fastest LDS layout.
    for (int t2 = tid; t2 < 16 * 198; t2 += 128) {
      const int icl = t2 / 198, r2 = t2 % 198, row = r2 / 66, col = r2 % 66;
      const int gh = h + row - 1, gw = col - 1;
      float v = 0.f;
      if (gh >= 0 && gh < 64 && gw >= 0 && gw < 64)
        v = x[((n * DIMC + icb * 16 + icl) * 64 + gh) * 64 + gw];
      slab[r2 * 16 + icl] = v;
    }
    if (icb + 1 < 16)  // hint the next slab into cache (global_prefetch_b8)
      __builtin_prefetch(&x[((n * DIMC + (icb + 1) * 16 + (tid & 15)) * 64 + h) * 64 +
                            ((tid >> 4) << 3)], 0, 0);
    __syncthreads();

    for (int kh = 0; kh < 3; ++kh)
      for (int kw = 0; kw < 3; ++kw) {
        const float* wrow = &wp[((ocbase + lane) * 9 + kh * 3 + kw) * DIMC + icb * 16 + koff];
        const float* srow = &slab[(kh * 66) * 16];
        #pragma unroll
        for (int k4 = 0; k4 < 4; ++k4) {
          v2f a = *(const v2f*)&wrow[k4 * 4];                 // 1x global b64
          const int icl = k4 * 4 + koff;
          #pragma unroll
          for (int nt = 0; nt < 4; ++nt) {
            const int col = nt * 16 + lane + kw;              // gw = w+(kw-1)
            v2f b = *(const v2f*)&srow[col * 16 + icl];       // 1x ds b64
            acc[nt] = wmma_f32(a, b, acc[nt]);
          }
        }
      }
  }
  #pragma unroll
  for (int nt = 0; nt < 4; ++nt)
    #pragma unroll
    for (int r = 0; r < 8; ++r) {
      const int oc = ocbase + r + 8 * hh;
      y0[((n * DIMC + oc) * 64 + h) * 64 + nt * 16 + lane] = acc[nt][r] + bc[oc];
    }
}

// ---------------------------------------------------------------------------
// Kernel 2: multi-scale separable depthwise (7/11/21) + identity, all in LDS.
// One (n,c) 64x64 plane per 256-thread block. s = x + sum_scales V(H(x)).
// ---------------------------------------------------------------------------
__global__ void __launch_bounds__(256)
k_dwms(const float* __restrict__ y0,
       const float* __restrict__ w01, const float* __restrict__ b01,
       const float* __restrict__ w02, const float* __restrict__ b02,
       const float* __restrict__ w11, const float* __restrict__ b11,
       const float* __restrict__ w12, const float* __restrict__ b12,
       const float* __restrict__ w21, const float* __restrict__ b21,
       const float* __restrict__ w22, const float* __restrict__ b22,
       float* __restrict__ sout) {
  __shared__ float plane[HW];
  __shared__ float tmp[HW];
  const int bid = blockIdx.x, n = bid >> 8, c = bid & 255, tid = threadIdx.x;
  const float* base = &y0[(long long)(n * DIMC + c) * HW];

  float acc[16];
  for (int i = 0; i < 16; ++i) plane[tid + 256 * i] = base[tid + 256 * i];
  __syncthreads();
  for (int i = 0; i < 16; ++i) acc[i] = plane[tid + 256 * i];   // identity term

  const float* wH[3] = {w01, w11, w21};
  const float* wV[3] = {w02, w12, w22};
  const float* bH[3] = {b01, b11, b21};
  const float* bV[3] = {b02, b12, b22};
  const int Ls[3] = {7, 11, 21};

  for (int sc = 0; sc < 3; ++sc) {
    const int L = Ls[sc], R = L >> 1;
    float wh[21], wvv[21];
    for (int j = 0; j < L; ++j) { wh[j] = wH[sc][c * L + j]; wvv[j] = wV[sc][c * L + j]; }
    const float bh = bH[sc][c], bv = bV[sc][c];
    for (int i = 0; i < 16; ++i) {                 // horizontal pass (zero pad)
      const int p = tid + 256 * i, hI = p >> 6, wI = p & 63;
      float sH = bh;
      for (int j = 0; j < L; ++j) {
        const int w2 = wI + j - R;
        if (w2 >= 0 && w2 < 64) sH += wh[j] * plane[(hI << 6) + w2];
      }
      tmp[p] = sH;
    }
    __syncthreads();
    for (int i = 0; i < 16; ++i) {                 // vertical pass (zero pad)
      const int p = tid + 256 * i, hI = p >> 6, wI = p & 63;
      float sV = bv;
      for (int j = 0; j < L; ++j) {
        const int h2 = hI + j - R;
        if (h2 >= 0 && h2 < 64) sV += wvv[j] * tmp[(h2 << 6) + wI];
      }
      acc[i] += sV;
    }
    __syncthreads();
  }
  float* outp = &sout[(long long)(n * DIMC + c) * HW];
  for (int i = 0; i < 16; ++i) outp[tid + 256 * i] = acc[i];
}

// ---------------------------------------------------------------------------
// Kernel 3: generic channel-wise GEMM (1x1 conv): out = W @ in (+bias)(+res).
// in: [B][K][4096], W: [M][K] (optionally per-batch), out: [B][M][4096].
// Grid: B*(M/16)*16 blocks of 128; each wave = 16 M x 64 pixels.
// ---------------------------------------------------------------------------
__global__ void __launch_bounds__(128)
k_gemm_cw(const float* __restrict__ in, long long in_bs,
          const float* __restrict__ W, long long w_bs,
          const float* __restrict__ bias,
          const float* __restrict__ res, long long res_bs,
          float* __restrict__ out, long long out_bs,
          int M, int K) {
  const int tid = threadIdx.x;
  const int lane = tid & 15, hh = (tid >> 4) & 1, wv = tid >> 5;
  const int wid = blockIdx.x * 4 + wv;
  const int perb = (M >> 4) * 64;
  const int b = wid / perb, r0 = wid % perb;
  const int mt = r0 >> 6, nbase = (r0 & 63) * 64;
  const float* Wb  = W  + (long long)b * w_bs;
  const float* inb = in + (long long)b * in_bs;

  v8f acc[4] = {};
  for (int kb = 0; kb < (K >> 2); ++kb) {
    const int k0 = kb * 4 + hh * 2;
    v2f a = *(const v2f*)&Wb[(mt * 16 + lane) * K + k0];
    const float* ip = &inb[(long long)k0 * HW + nbase + lane];
    #pragma unroll
    for (int nt = 0; nt < 4; ++nt) {
      v2f bf; bf.x = ip[nt * 16]; bf.y = ip[nt * 16 + HW];
      acc[nt] = wmma_f32(a, bf, acc[nt]);
    }
  }
  #pragma unroll
  for (int nt = 0; nt < 4; ++nt)
    #pragma unroll
    for (int r = 0; r < 8; ++r) {
      const int row = mt * 16 + r + 8 * hh;
      const int p = nbase + nt * 16 + lane;
      float v = acc[nt][r];
      if (bias) v += bias[row];
      if (res)  v += res[(long long)b * res_bs + (long long)row * HW + p];
      out[(long long)b * out_bs + (long long)row * HW + p] = v;
    }
}

// ---------------------------------------------------------------------------
// Kernel 4: channel attention weights: G = t t^T (WMMA, K=4096), then
// S_ij = G_ij / (max(sqrt(G_ii),eps) * max(sqrt(G_jj),eps)), row softmax.
// One block (4 waves = 4 G-tiles) per batch.
// ---------------------------------------------------------------------------
__global__ void __launch_bounds__(128)
k_chan_attn(const float* __restrict__ t, float* __restrict__ attn1) {
  __shared__ float G[32 * 32];
  __shared__ float nrm[32];
  const int b = blockIdx.x, tid = threadIdx.x;
  const int lane = tid & 15, hh = (tid >> 4) & 1, wv = tid >> 5;
  const int mt = wv >> 1, nt = wv & 1;
  const float* tb = &t[(long long)b * TSTRIDE];

  v8f acc = {};
  for (int kb = 0; kb < 1024; ++kb) {
    const int k0 = kb * 4 + hh * 2;
    v2f a  = *(const v2f*)&tb[(mt * 16 + lane) * HW + k0];
    v2f bf = *(const v2f*)&tb[(nt * 16 + lane) * HW + k0];
    acc = wmma_f32(a, bf, acc);
  }
  #pragma unroll
  for (int r = 0; r < 8; ++r)
    G[(mt * 16 + r + 8 * hh) * 32 + nt * 16 + lane] = acc[r];
  __syncthreads();
  if (tid < 32) nrm[tid] = fmaxf(sqrtf(fmaxf(G[tid * 33], 0.f)), 1e-12f);
  __syncthreads();
  if (tid < 32) {
    float sv[32], mx = -1e30f;
    for (int j = 0; j < 32; ++j) {
      sv[j] = G[tid * 32 + j] / (nrm[tid] * nrm[j]);
      mx = fmaxf(mx, sv[j]);
    }
    float sum = 0.f;
    for (int j = 0; j < 32; ++j) { sv[j] = __expf(sv[j] - mx); sum += sv[j]; }
    const float inv = 1.f / sum;
    for (int j = 0; j < 32; ++j) attn1[b * 1024 + tid * 32 + j] = sv[j] * inv;
  }
}

// ---------------------------------------------------------------------------
// Kernel 5a: q2 = row-l2norm of ts = t^T  -> q2[b][p][0..31] contiguous.
// ---------------------------------------------------------------------------
__global__ void __launch_bounds__(256)
k_q2norm(const float* __restrict__ t, float* __restrict__ q2) {
  const int b = blockIdx.x >> 4;
  const int p = ((blockIdx.x & 15) << 8) + threadIdx.x;
  const float* tb = &t[(long long)b * TSTRIDE];
  float v[32], ss = 0.f;
  for (int c = 0; c < 32; ++c) { v[c] = tb[c * HW + p]; ss += v[c] * v[c]; }
  const float inv = 1.f / fmaxf(sqrtf(ss), 1e-12f);
  float* qp = &q2[(long long)b * TSTRIDE + (long long)p * 32];
  for (int c = 0; c < 32; ++c) qp[c] = v[c] * inv;
}

// ---------------------------------------------------------------------------
// Kernel 5b: flash spatial attention: out3 = (softmax(q2 q2^T) @ ts)^T,
// streamed over 256 KV tiles of 16; never materializes the 4096x4096 matrix.
// One wave = 16 query rows. P round-trips through per-wave LDS to re-shape
// the softmax tile (C/D layout) into an A-fragment; same-wave LDS ordering is
// enforced with s_wait_dscnt 0.  Writes rows 32..63 of out4.
// ---------------------------------------------------------------------------
__global__ void __launch_bounds__(128)
k_flash_spatial(const float* __restrict__ q2, const float* __restrict__ t,
                float* __restrict__ out4) {
  __shared__ float P[4 * 256];
  const int tid = threadIdx.x;
  const int lane = tid & 15, hh = (tid >> 4) & 1, wv = tid >> 5;
  const int b = blockIdx.x >> 6;
  const int qbase = ((blockIdx.x & 63) * 4 + wv) * 16;
  const float* qb = &q2[(long long)b * TSTRIDE];
  const float* tb = &t[(long long)b * TSTRIDE];
  float* Pw = &P[wv * 256];

  v2f qa[8];
  #pragma unroll
  for (int kk = 0; kk < 8; ++kk)
    qa[kk] = *(const v2f*)&qb[(qbase + lane) * 32 + kk * 4 + hh * 2];

  v8f o0 = {}, o1 = {};
  float mrow[8], lrow[8];
  #pragma unroll
  for (int r = 0; r < 8; ++r) { mrow[r] = -1e30f; lrow[r] = 0.f; }

  for (int kv = 0; kv < 256; ++kv) {
    const int kvb = kv * 16;
    v8f sacc = {};
    #pragma unroll
    for (int kk = 0; kk < 8; ++kk) {                       // S = Q K^T (K=32)
      v2f kb = *(const v2f*)&qb[(kvb + lane) * 32 + kk * 4 + hh * 2];
      sacc = wmma_f32(qa[kk], kb, sacc);
    }
    #pragma unroll
    for (int r = 0; r < 8; ++r) {                          // online softmax
      const float v = sacc[r];
      float rmax = v;
      for (int msk = 8; msk >= 1; msk >>= 1) rmax = fmaxf(rmax, __shfl_xor(rmax, msk, 16));
      const float mnew = fmaxf(mrow[r], rmax);
      const float pe = __expf(v - mnew);
      float rsum = pe;
      for (int msk = 8; msk >= 1; msk >>= 1) rsum += __shfl_xor(rsum, msk, 16);
      const float alpha = __expf(mrow[r] - mnew);
      lrow[r] = lrow[r] * alpha + rsum;
      mrow[r] = mnew;
      o0[r] *= alpha; o1[r] *= alpha;
      Pw[(r + 8 * hh) * 16 + lane] = pe;                   // store P tile
    }
    asm volatile("s_wait_dscnt 0" ::: "memory");           // wave-local LDS RAW
    #pragma unroll
    for (int kk2 = 0; kk2 < 4; ++kk2) {                    // O += P V (K=16)
      v2f pa = *(const v2f*)&Pw[lane * 16 + kk2 * 4 + hh * 2];
      const int kcol = kvb + kk2 * 4 + hh * 2;
      v2f vb0 = *(const v2f*)&tb[lane * HW + kcol];
      v2f vb1 = *(const v2f*)&tb[(16 + lane) * HW + kcol];
      o0 = wmma_f32(pa, vb0, o0);
      o1 = wmma_f32(pa, vb1, o1);
    }
  }
  float* ob = &out4[(long long)b * (64LL * HW)];
  #pragma unroll
  for (int r = 0; r < 8; ++r) {
    const int prow = qbase + r + 8 * hh;
    const float inv = 1.f / lrow[r];
    ob[(32 + lane) * HW + prow] = o0[r] * inv;             // out3 channels 0..15
    ob[(48 + lane) * HW + prow] = o1[r] * inv;             // out3 channels 16..31
  }
}

// ---------------------------------------------------------------------------
// Host-side orchestration.
// ---------------------------------------------------------------------------
extern "C" void kernel_launch(void* const* d_in, const int* in_sizes, int n_in,
                              void* d_out, int out_size, void* d_ws, size_t ws_size,
                              hipStream_t stream) {
  (void)in_sizes; (void)n_in; (void)out_size; (void)ws_size;
  const float* x      = (const float*)d_in[0];
  const float* w_conv = (const float*)d_in[1];
  const float* b_conv = (const float*)d_in[2];
  const float* w01 = (const float*)d_in[3];  const float* b01 = (const float*)d_in[4];
  const float* w02 = (const float*)d_in[5];  const float* b02 = (const float*)d_in[6];
  const float* w11 = (const float*)d_in[7];  const float* b11 = (const float*)d_in[8];
  const float* w12 = (const float*)d_in[9];  const float* b12 = (const float*)d_in[10];
  const float* w21 = (const float*)d_in[11]; const float* b21 = (const float*)d_in[12];
  const float* w22 = (const float*)d_in[13]; const float* b22 = (const float*)d_in[14];
  const float* w_proj = (const float*)d_in[15]; const float* b_proj = (const float*)d_in[16];
  const float* w3 = (const float*)d_in[17]; const float* b3 = (const float*)d_in[18];
  const float* w4 = (const float*)d_in[19]; const float* b4 = (const float*)d_in[20];

  float* ws = (float*)d_ws;
  // workspace layout (floats): total 11,603,968 floats = 46.4 MB
  float* y0    = ws + 0;         // [4,256,4096] conv3x3 out
  float* s     = ws + 4194304;   // [4,256,4096] multi-scale sum
  float* t     = ws + 8388608;   // [4, 32,4096] out1
  float* q2    = ws + 8912896;   // [4,4096, 32] l2norm(ts)
  float* attn1 = ws + 9437184;   // [4, 32,  32]
  float* out4  = ws + 9441280;   // [4, 64,4096] concat(out2,out3)
  float* out5  = ws + 10489856;  // [4, 32,4096]
  float* wpk   = ws + 11014144;  // [256,9,256] repacked conv weights

  // 0) repack conv weights for b64 A-fragment loads
  k_repack_w<<<2304, 256, 0, stream>>>(w_conv, wpk);
  // 1) dense 3x3 conv (WMMA implicit GEMM)
  k_conv3x3<<<1024, 128, 0, stream>>>(x, wpk, b_conv, y0);
  // 2) multi-scale separable depthwise + identity
  k_dwms<<<1024, 256, 0, stream>>>(y0, w01, b01, w02, b02, w11, b11,
                                   w12, b12, w21, b21, w22, b22, s);
  // 3) 1x1 projection 256->32 : t = out1
  k_gemm_cw<<<128, 128, 0, stream>>>(s, 256LL * HW, w_proj, 0, b_proj,
                                     nullptr, 0, t, TSTRIDE, 32, 256);
  // 4) channel attention: attn1 then out2 -> out4 rows 0..31
  k_chan_attn<<<4, 128, 0, stream>>>(t, attn1);
  k_gemm_cw<<<128, 128, 0, stream>>>(t, TSTRIDE, attn1, 1024, nullptr,
                                     nullptr, 0, out4, 64LL * HW, 32, 32);
  // 5) spatial attention (flash) -> out4 rows 32..63
  k_q2norm<<<64, 256, 0, stream>>>(t, q2);
  k_flash_spatial<<<256, 128, 0, stream>>>(q2, t, out4);
  // 6) out5 = w3 @ out4 + b3 + out1 ;  out = w4 @ out5 + b4
  k_gemm_cw<<<128, 128, 0, stream>>>(out4, 64LL * HW, w3, 0, b3,
                                     t, TSTRIDE, out5, TSTRIDE, 32, 64);
  k_gemm_cw<<<1024, 128, 0, stream>>>(out5, TSTRIDE, w4, 0, b4,
                                      nullptr, 0, (float*)d_out, 256LL * HW, 256, 32);
}